// KeyPointNet_mod_76544907149601
// MI455X (gfx1250) — compile-verified
//
#include <hip/hip_runtime.h>

typedef float v2f __attribute__((ext_vector_type(2)));
typedef float v8f __attribute__((ext_vector_type(8)));
typedef unsigned int u32x4 __attribute__((ext_vector_type(4)));
typedef unsigned int u32x8 __attribute__((ext_vector_type(8)));

#define N_PTS   4096
#define KNN     20
#define NKEY    512
#define EMB_D   512
#define WAVES   4            // waves per sharpness block

// ---------------------------------------------------------------------------
// Kernel 1: per-point KNN covariance sharpness.
//  * Points staged into LDS once per block by the Tensor Data Mover
//    (tensor_load_to_lds, TENSORcnt) in the global SoA [3][4096] layout.
//  * Pairwise-distance Gram tiles via V_WMMA_F32_16X16X4_F32 (K=3 padded to 4).
//  * 4 wave32 per block, each owning one 16-row tile; lane owns a fixed row
//    (C-matrix N dim), keeps a sorted top-20 list in VGPRs over its 8-column
//    half of every tile; the lane pair (l, l+16) unions its lists by rank
//    counting (no dynamic register indexing) + __shfl_down reduction.
// ---------------------------------------------------------------------------
__global__ __launch_bounds__(WAVES * 32) void sharpness_kernel(
    const float* __restrict__ src, const float* __restrict__ tgt,
    float* __restrict__ sharp /* [2,B,N] */, int B)
{
    const int sel = blockIdx.y & 1;      // 0 = src, 1 = tgt
    const int b   = blockIdx.y >> 1;
    const float* __restrict__ P = (sel ? tgt : src) + (size_t)b * 3 * N_PTS;

    const int lane = threadIdx.x & 31;
    const int wid  = threadIdx.x >> 5;
    const int rloc = lane & 15;
    const int half = lane >> 4;
    const int tile = blockIdx.x * WAVES + wid;
    const int r    = tile * 16 + rloc;            // this lane's home row

    // LDS: points in global SoA layout [3][N] (TDM target) + squared norms.
    // The norm region is re-used as the top-K distance-list buffer at the end.
    __shared__ float smem[3 * N_PTS + N_PTS];     // 64 KB, LDS offset 0
    float* ptsL = smem;                           // [3*N_PTS]
    float* sqL  = smem + 3 * N_PTS;               // [N_PTS]

    // ---- TDM preload: one wave issues a 2D tensor load [3 x 4096] f32 ----
    if (threadIdx.x < 32) {
        const unsigned long long ga = (unsigned long long)(size_t)P;
        // D# group 0: count=1 | lds_addr=0 | global_addr[56:0] | type=2
        u32x4 g0;
        g0[0] = 1u;                               // count=1, user mode
        g0[1] = 0u;                               // lds_addr = 0 (smem base)
        g0[2] = (unsigned int)ga;                 // global_addr[31:0]
        g0[3] = ((unsigned int)(ga >> 32) & 0x01FFFFFFu) | (2u << 30); // type=2
        // D# group 1: data_size=4B, dims 4096 x 3, tile 4096 x 3, stride 4096
        u32x8 g1;
        g1[0] = 2u << 16;                         // data_size = 2 (4 bytes)
        g1[1] = (unsigned int)N_PTS << 16;        // tensor_dim0[15:0]
        g1[2] = 3u << 16;                         // dim0 hi=0 | tensor_dim1=3
        g1[3] = (unsigned int)N_PTS << 16;        // dim1 hi=0 | tile_dim0=4096
        g1[4] = 3u;                               // tile_dim1=3 | tile_dim2=0
        g1[5] = (unsigned int)N_PTS;              // tensor_dim0_stride lo
        g1[6] = 0u;                               // stride hi | dim1_stride lo
        g1[7] = 0u;
        asm volatile("tensor_load_to_lds %0, %1" :: "s"(g0), "s"(g1) : "memory");
        __builtin_amdgcn_s_wait_tensorcnt(0);
    }
    __syncthreads();

    // squared norms from LDS
    for (int n = threadIdx.x; n < N_PTS; n += WAVES * 32) {
        const float x = ptsL[n];
        const float y = ptsL[N_PTS + n];
        const float z = ptsL[2 * N_PTS + n];
        sqL[n] = x * x + y * y + z * z;
    }
    __syncthreads();

    const float px  = ptsL[r];
    const float py  = ptsL[N_PTS + r];
    const float pz  = ptsL[2 * N_PTS + r];
    const float psq = sqL[r];

    // B fragment (4x16): lanes 0-15: VGPR0=K0,VGPR1=K1; 16-31: K2,K3(=0 pad)
    v2f bfrag;
    bfrag.x = half ? pz : px;
    bfrag.y = half ? 0.0f : py;

    float kd[KNN];
    unsigned int ki[KNN];
#pragma unroll
    for (int i = 0; i < KNN; ++i) { kd[i] = 3.0e38f; ki[i] = 0u; }

    for (int ct = 0; ct < N_PTS / 16; ++ct) {
        const int c = ct * 16 + rloc;
        v2f afrag;                                 // A fragment (16x4)
        {
            const float qx = ptsL[c];
            const float qy = ptsL[N_PTS + c];
            const float qz = ptsL[2 * N_PTS + c];
            afrag.x = half ? qz : qx;
            afrag.y = half ? 0.0f : qy;
        }
        v8f acc = {};
        acc = __builtin_amdgcn_wmma_f32_16x16x4_f32(
            false, afrag, false, bfrag, (short)0, acc, false, false);

#pragma unroll
        for (int v = 0; v < 8; ++v) {
            const int m = ct * 16 + v + 8 * half;  // candidate point index
            const float d2 = sqL[m] + psq - 2.0f * acc[v];
            if (d2 < kd[KNN - 1]) {                // sorted insertion (VGPRs)
                kd[KNN - 1] = d2;
                ki[KNN - 1] = (unsigned int)m;
#pragma unroll
                for (int s = KNN - 1; s > 0; --s) {
                    if (kd[s] < kd[s - 1]) {
                        float td = kd[s]; kd[s] = kd[s - 1]; kd[s - 1] = td;
                        unsigned int ti = ki[s]; ki[s] = ki[s - 1]; ki[s - 1] = ti;
                    }
                }
            }
        }
    }
    __syncthreads();       // all waves done with sqL -> reuse it below

    // ---- union of the two 20-lists per row by rank counting --------------
    // Each lane publishes its 20 distances (only) to LDS.
    float* distL = sqL + (size_t)wid * (32 * KNN);      // 640 floats / wave
#pragma unroll
    for (int i = 0; i < KNN; ++i) distL[lane * KNN + i] = kd[i];
    __syncthreads();

    const float* pd = distL + (lane ^ 16) * KNN;        // partner's list
    float sx = 0, sy = 0, sz = 0;
    float sxx = 0, syy = 0, szz = 0, sxy = 0, sxz = 0, syz = 0;
#pragma unroll
    for (int i = 0; i < KNN; ++i) {
        const float d = kd[i];
        int cnt = 0;
#pragma unroll
        for (int j = 0; j < KNN; ++j) {
            const float w = pd[j];
            // strict total order: ties go to the lower lane (half==0)
            cnt += (w < d) || (half && (w == d));
        }
        if (i + cnt < KNN) {                      // in the merged top-20
            const unsigned int id = ki[i];
            const float x = ptsL[id];
            const float y = ptsL[N_PTS + id];
            const float z = ptsL[2 * N_PTS + id];
            sx += x; sy += y; sz += z;
            sxx += x * x; syy += y * y; szz += z * z;
            sxy += x * y; sxz += x * z; syz += y * z;
        }
    }
    // combine the lane pair (l gets l+16's partials)
    sx  += __shfl_down(sx, 16, 32);   sy  += __shfl_down(sy, 16, 32);
    sz  += __shfl_down(sz, 16, 32);   sxx += __shfl_down(sxx, 16, 32);
    syy += __shfl_down(syy, 16, 32);  szz += __shfl_down(szz, 16, 32);
    sxy += __shfl_down(sxy, 16, 32);  sxz += __shfl_down(sxz, 16, 32);
    syz += __shfl_down(syz, 16, 32);

    if (lane < 16) {
        const float invk = 1.0f / (float)KNN;
        const float mx = sx * invk, my = sy * invk, mz = sz * invk;
        const float cxx = sxx * invk - mx * mx;
        const float cyy = syy * invk - my * my;
        const float czz = szz * invk - mz * mz;
        const float cxy = sxy * invk - mx * my;
        const float cxz = sxz * invk - mx * mz;
        const float cyz = syz * invk - my * mz;
        const float det = cxx * (cyy * czz - cyz * cyz)
                        - cxy * (cxy * czz - cyz * cxz)
                        + cxz * (cxy * cyz - cyy * cxz);
        const float tr = cxx + cyy + czz;
        sharp[((size_t)sel * B + b) * N_PTS + r] = fmaxf(det, 0.0f) - cbrtf(tr);
    }
}

// ---------------------------------------------------------------------------
// Kernel 2: stable top-512 by rank counting (descending value, index tiebreak),
// matching jax.lax.top_k ordering. One 256-thread block per (sel,b) row.
// ---------------------------------------------------------------------------
__global__ __launch_bounds__(256) void select_kernel(
    const float* __restrict__ sharp, int* __restrict__ idxbuf, int B)
{
    const int row = blockIdx.x;                 // sel*B + b
    const float* __restrict__ s = sharp + (size_t)row * N_PTS;
    __shared__ float sL[N_PTS];
    for (int n = threadIdx.x; n < N_PTS; n += 256) sL[n] = s[n];
    __syncthreads();

    for (int n = threadIdx.x; n < N_PTS; n += 256) {
        const float v = sL[n];
        int rank = 0;
        for (int m = 0; m < N_PTS; ++m) {
            const float w = sL[m];
            rank += (w > v) || (w == v && m < n);
        }
        if (rank < NKEY) idxbuf[(size_t)row * NKEY + rank] = n;
    }
}

// ---------------------------------------------------------------------------
// Kernel 3: gathers. Writes fully coalesced; reads hit L2 (everything fits).
// d_out layout: src_kp[B,3,C] | tgt_kp[B,3,C] | src_emb[B,D,C] | tgt_emb[B,D,C]
// ---------------------------------------------------------------------------
__global__ __launch_bounds__(256) void gather_kernel(
    const float* __restrict__ src, const float* __restrict__ tgt,
    const float* __restrict__ semb, const float* __restrict__ temb,
    const int* __restrict__ idxbuf, float* __restrict__ out, int B)
{
    const size_t kpSz  = (size_t)B * 3 * NKEY;
    const size_t embSz = (size_t)B * EMB_D * NKEY;
    const size_t total = 2 * kpSz + 2 * embSz;
    const size_t tid = (size_t)blockIdx.x * blockDim.x + threadIdx.x;
    if (tid >= total) return;

    if (tid < 2 * kpSz) {
        const int which = tid >= kpSz;
        size_t e = tid - (size_t)which * kpSz;
        const int b = (int)(e / (3 * NKEY));
        const int d = (int)((e / NKEY) % 3);
        const int c = (int)(e % NKEY);
        const int idx = idxbuf[((size_t)which * B + b) * NKEY + c];
        const float* __restrict__ Pp = which ? tgt : src;
        out[tid] = Pp[((size_t)b * 3 + d) * N_PTS + idx];
    } else {
        size_t e = tid - 2 * kpSz;
        const int which = e >= embSz;
        e -= (size_t)which * embSz;
        const int b = (int)(e / ((size_t)EMB_D * NKEY));
        const int d = (int)((e / NKEY) % EMB_D);
        const int c = (int)(e % NKEY);
        const int idx = idxbuf[((size_t)which * B + b) * NKEY + c];
        const float* __restrict__ E = which ? temb : semb;
        out[tid] = E[((size_t)b * EMB_D + d) * N_PTS + idx];
    }
}

// ---------------------------------------------------------------------------
extern "C" void kernel_launch(void* const* d_in, const int* in_sizes, int n_in,
                              void* d_out, int out_size, void* d_ws, size_t ws_size,
                              hipStream_t stream)
{
    const float* src  = (const float*)d_in[0];
    const float* tgt  = (const float*)d_in[1];
    const float* semb = (const float*)d_in[2];
    const float* temb = (const float*)d_in[3];

    const int B = in_sizes[0] / (3 * N_PTS);    // = 4

    float* sharp  = (float*)d_ws;                                   // [2,B,N]
    int*   idxbuf = (int*)((char*)d_ws + sizeof(float) * 2 * (size_t)B * N_PTS);

    dim3 g1(N_PTS / (16 * WAVES), 2 * B);
    sharpness_kernel<<<g1, WAVES * 32, 0, stream>>>(src, tgt, sharp, B);

    select_kernel<<<2 * B, 256, 0, stream>>>(sharp, idxbuf, B);

    const size_t total = 2 * (size_t)B * 3 * NKEY + 2 * (size_t)B * EMB_D * NKEY;
    gather_kernel<<<(unsigned)((total + 255) / 256), 256, 0, stream>>>(
        src, tgt, semb, temb, idxbuf, (float*)d_out, B);
}